// MultiHeadGATLayerWithEdgeFeatures_42949672960527
// MI455X (gfx1250) — compile-verified
//
#include <hip/hip_runtime.h>
#include <hip/hip_bf16.h>
#include <math.h>

#define N_NODES 50000
#define E_EDGES 800000
#define F_INDIM 128
#define DDIM    32
#define HEADS   4
#define FEDIM   16
#define F_OUT   128   // HEADS * DDIM
#define ALPHA   0.2f

typedef float v2f __attribute__((ext_vector_type(2)));
typedef float v8f __attribute__((ext_vector_type(8)));

// ---------------------------------------------------------------------------
// Kernel 0: zero the output accumulator (segment_sum target).
// ---------------------------------------------------------------------------
__global__ __launch_bounds__(256) void zero_out_kernel(float4* __restrict__ out) {
    const int i = blockIdx.x * blockDim.x + threadIdx.x;   // 1.6M float4s exactly
    out[i] = make_float4(0.f, 0.f, 0.f, 0.f);
}

// ---------------------------------------------------------------------------
// Kernel 1: Wh = h @ W  via V_WMMA_F32_16X16X4_F32.
// One wave computes one 16x16 f32 tile of Wh; K=128 stepped by 4.
// A (16x4 f32): lanes 0-15 hold rows M=0..15 with K={k,k+1} in VGPR0/1,
//               lanes 16-31 hold K={k+2,k+3}.
// B (4x16 f32): lanes 0-15 hold cols N with K={k,k+1} in VGPR0/1,
//               lanes 16-31 hold K={k+2,k+3}.
// C/D (16x16 f32): VGPR r -> row M=r (lanes 0-15) / M=8+r (lanes 16-31).
// ---------------------------------------------------------------------------
__global__ __launch_bounds__(128) void gemm_wmma_kernel(
        const float* __restrict__ h, const float* __restrict__ W,
        float* __restrict__ Wh) {
    const int lane   = threadIdx.x & 31;
    const int wave   = blockIdx.x * (blockDim.x >> 5) + (threadIdx.x >> 5);
    const int rowT   = wave >> 3;          // 3125 row tiles
    const int colT   = wave & 7;           // 8 col tiles (128/16)
    const int row0   = rowT * 16;
    const int col0   = colT * 16;
    const int half   = lane >> 4;          // 0: K+0/1, 1: K+2/3
    const int lane16 = lane & 15;

    v8f c = {0.f, 0.f, 0.f, 0.f, 0.f, 0.f, 0.f, 0.f};

    const float* arow = h + (size_t)(row0 + lane16) * F_INDIM;
    #pragma unroll 4
    for (int k = 0; k < F_INDIM; k += 4) {
        const int kk = k + half * 2;
        // A fragment: two consecutive K values for this lane's row
        const float2 af = *(const float2*)(arow + kk);
        v2f a; a.x = af.x; a.y = af.y;
        // B fragment: same two K rows of W at this lane's column
        const float* bp = W + (size_t)kk * F_OUT + col0 + lane16;
        v2f b; b.x = bp[0]; b.y = bp[F_OUT];
        c = __builtin_amdgcn_wmma_f32_16x16x4_f32(
                /*neg_a=*/false, a, /*neg_b=*/false, b,
                /*c_mod=*/(short)0, c, /*reuse_a=*/false, /*reuse_b=*/false);
    }

    const int col = col0 + lane16;
    #pragma unroll
    for (int r = 0; r < 8; ++r) {
        const int row = row0 + half * 8 + r;
        Wh[(size_t)row * F_OUT + col] = c[r];
    }
}

// ---------------------------------------------------------------------------
// Kernel 2: per-edge attention + scatter. One wave32 per edge.
// Lane l owns features [4l, 4l+4) -> head = l/8, dim offset = (4l)&31.
// Logit reduction over the 8 lanes of each head via shfl_xor; softmax over
// the 4 heads via lane broadcasts. Message scattered with f32 HW atomics.
// ---------------------------------------------------------------------------
__global__ __launch_bounds__(256) void edge_kernel(
        const float* __restrict__ Wh, const float* __restrict__ edge_attr,
        const float* __restrict__ a, const int* __restrict__ edge_index,
        float* __restrict__ out) {
    const int lane = threadIdx.x & 31;
    const int e    = (blockIdx.x * blockDim.x + threadIdx.x) >> 5;  // exactly E
    const int src  = edge_index[e];
    const int dst  = edge_index[E_EDGES + e];

    const float4 wi = ((const float4*)(Wh + (size_t)src * F_OUT))[lane];
    const float4 wj = ((const float4*)(Wh + (size_t)dst * F_OUT))[lane];

    const int head = lane >> 3;
    const int doff = (lane * 4) & (DDIM - 1);
    const float* ah = a + head * (2 * DDIM + FEDIM);   // row of `a` for this head

    // e_h = Wh_i . a_i + Wh_j . a_j  (partial over this lane's 4 dims)
    float p = wi.x * ah[doff]            + wi.y * ah[doff + 1]
            + wi.z * ah[doff + 2]        + wi.w * ah[doff + 3];
    p      += wj.x * ah[DDIM + doff]     + wj.y * ah[DDIM + doff + 1]
            + wj.z * ah[DDIM + doff + 2] + wj.w * ah[DDIM + doff + 3];

    // + edge_attr . a_e : lane-in-group g covers fe = {2g, 2g+1}
    const int g = lane & 7;
    const float2 ea = ((const float2*)(edge_attr + (size_t)e * FEDIM))[g];
    p += ea.x * ah[2 * DDIM + 2 * g] + ea.y * ah[2 * DDIM + 2 * g + 1];

    // reduce across the 8 lanes of this head
    p += __shfl_xor(p, 1, 32);
    p += __shfl_xor(p, 2, 32);
    p += __shfl_xor(p, 4, 32);

    // gather all 4 head logits, LeakyReLU, softmax over heads
    float e0 = __shfl(p, 0, 32), e1 = __shfl(p, 8, 32);
    float e2 = __shfl(p, 16, 32), e3 = __shfl(p, 24, 32);
    e0 = e0 > 0.f ? e0 : ALPHA * e0;
    e1 = e1 > 0.f ? e1 : ALPHA * e1;
    e2 = e2 > 0.f ? e2 : ALPHA * e2;
    e3 = e3 > 0.f ? e3 : ALPHA * e3;
    const float m  = fmaxf(fmaxf(e0, e1), fmaxf(e2, e3));
    const float x0 = __expf(e0 - m), x1 = __expf(e1 - m);
    const float x2 = __expf(e2 - m), x3 = __expf(e3 - m);
    const float inv = 1.f / (x0 + x1 + x2 + x3);
    const float own = (head == 0) ? x0 : (head == 1) ? x1 : (head == 2) ? x2 : x3;
    const float att = own * inv;

    // scatter: out[src] += att * Wh_j   (L2-resident f32 atomics)
    float* dp = out + (size_t)src * F_OUT + lane * 4;
    unsafeAtomicAdd(dp + 0, att * wj.x);
    unsafeAtomicAdd(dp + 1, att * wj.y);
    unsafeAtomicAdd(dp + 2, att * wj.z);
    unsafeAtomicAdd(dp + 3, att * wj.w);
}

// ---------------------------------------------------------------------------
// Kernel 3: in-place ELU on the accumulated output.
// ---------------------------------------------------------------------------
__global__ __launch_bounds__(256) void elu_kernel(float* __restrict__ out) {
    const int i = blockIdx.x * blockDim.x + threadIdx.x;   // 6.4M exactly
    const float x = out[i];
    out[i] = x > 0.f ? x : (__expf(x) - 1.f);
}

// ---------------------------------------------------------------------------
extern "C" void kernel_launch(void* const* d_in, const int* in_sizes, int n_in,
                              void* d_out, int out_size, void* d_ws, size_t ws_size,
                              hipStream_t stream) {
    const float* h          = (const float*)d_in[0];   // (N, 128)
    const float* edge_attr  = (const float*)d_in[1];   // (E, 16)
    const float* W          = (const float*)d_in[2];   // (128, 128)
    const float* a          = (const float*)d_in[3];   // (4, 80)
    const int*   edge_index = (const int*)d_in[4];     // (2, E)
    float* out = (float*)d_out;                        // (N, 128)
    float* Wh  = (float*)d_ws;                         // scratch: N*128 f32 = 25.6 MB

    // 1) zero the segment-sum accumulator (6.4M floats = 1.6M float4s)
    zero_out_kernel<<<(N_NODES * F_OUT / 4) / 256, 256, 0, stream>>>((float4*)out);

    // 2) Wh = h @ W : 3125*8 = 25000 wave-tiles, 4 waves per block
    gemm_wmma_kernel<<<(3125 * 8) / 4, 128, 0, stream>>>(h, W, Wh);

    // 3) per-edge attention + scatter: 1 wave per edge, 8 waves per block
    edge_kernel<<<E_EDGES / 8, 256, 0, stream>>>(Wh, edge_attr, a, edge_index, out);

    // 4) ELU in place
    elu_kernel<<<(N_NODES * F_OUT) / 256, 256, 0, stream>>>(out);
}